// GraphAttentionLayer_71725953843274
// MI455X (gfx1250) — compile-verified
//
#include <hip/hip_runtime.h>
#include <hip/hip_bf16.h>
#include <stdint.h>

// ---------------------------------------------------------------------------
// GAT layer for MI455X (gfx1250, wave32).
//   Wh = h @ W            : fp32 WMMA (v_wmma_f32_16x16x4_f32)
//   s1 = Wh@a1, s2 = Wh@a2: wave-per-row GEMV
//   edge softmax (segment max/sum over src) : ordered-uint atomic max + f32 atomic add
//   h' = segment_sum(att * Wh[dst]) : wave-per-edge float4 gather + native f32 atomics
//   relu in place
// ---------------------------------------------------------------------------

#define IN_F   512
#define OUT_F  128
#define NEG_SLOPE 0.01f
#define INV_TEMP  2.0f   // 1/0.5

typedef __attribute__((ext_vector_type(2))) float v2f;
typedef __attribute__((ext_vector_type(8))) float v8f;

// order-preserving monotonic mapping float -> uint (for atomic max on floats)
__device__ __forceinline__ unsigned f2ord(float f) {
  unsigned u = __float_as_uint(f);
  return (u & 0x80000000u) ? ~u : (u | 0x80000000u);
}
__device__ __forceinline__ float ord2f(unsigned u) {
  return __uint_as_float((u & 0x80000000u) ? (u ^ 0x80000000u) : ~u);
}
#define ORD_NEG_INF 0x007FFFFFu   // f2ord(-inf)

// native fp32 atomic add (global_atomic_add_f32, not a CAS loop)
__device__ __forceinline__ void atomAddF32(float* p, float v) {
  unsafeAtomicAdd(p, v);
}

// ---------------------------------------------------------------------------
// 0) init: zero out accumulator, denom; seg_max = -inf (ordered encoding)
// ---------------------------------------------------------------------------
__global__ __launch_bounds__(256) void gat_init(float* __restrict__ out,
                                                unsigned* __restrict__ segmax,
                                                float* __restrict__ denom,
                                                int n_nodes) {
  int i = blockIdx.x * blockDim.x + threadIdx.x;
  int total = n_nodes * OUT_F;
  if (i < total) out[i] = 0.0f;
  if (i < n_nodes) { segmax[i] = ORD_NEG_INF; denom[i] = 0.0f; }
}

// ---------------------------------------------------------------------------
// 1) Wh = h @ W using fp32 WMMA 16x16x4.
//    Block = 256 threads = 8 waves. Block computes a 16-row x 128-col tile;
//    each wave owns one 16x16 tile and runs K=512 -> 128 WMMA ops.
//    A-frag layout (16x4 f32): lane<16 holds A[row=lane][k, k+1],
//    lane>=16 holds A[row=lane-16][k+2, k+3]  (contiguous float2 load).
//    B-frag: v0 = B[k + 2*half][col], v1 = B[k + 2*half + 1][col].
//    C/D: vgpr i holds D[row0 + i + 8*half][col].
// ---------------------------------------------------------------------------
__global__ __launch_bounds__(256) void gat_gemm_wh(const float* __restrict__ h,
                                                   const float* __restrict__ W,
                                                   float* __restrict__ Wh,
                                                   int n_nodes) {
  const int lane = threadIdx.x & 31;
  const int wave = threadIdx.x >> 5;          // 0..7 -> column tile
  const int idx  = lane & 15;
  const int half = lane >> 4;                 // 0 or 1
  const int row0 = blockIdx.x * 16;
  const int col0 = wave * 16;

  int rowA = row0 + idx;
  if (rowA >= n_nodes) rowA = n_nodes - 1;    // clamp (keeps EXEC all-ones)
  const float* __restrict__ ha = h + (size_t)rowA * IN_F;
  const int colB = col0 + idx;

  v8f acc = {};
#pragma unroll 4
  for (int k = 0; k < IN_F; k += 4) {
    const int kk = k + 2 * half;
    v2f afrag;
    {   // contiguous 8-byte load: h[rowA][kk], h[rowA][kk+1]
      float2 t = *(const float2*)(ha + kk);
      afrag.x = t.x; afrag.y = t.y;
    }
    v2f bfrag;
    bfrag.x = W[(size_t)(kk    ) * OUT_F + colB];
    bfrag.y = W[(size_t)(kk + 1) * OUT_F + colB];
    acc = __builtin_amdgcn_wmma_f32_16x16x4_f32(
        /*neg_a=*/false, afrag, /*neg_b=*/false, bfrag,
        /*c_mod=*/(short)0, acc, /*reuse_a=*/false, /*reuse_b=*/false);
  }

  // Tile-uniform guard: one scalar branch instead of 8 EXEC-masked stores.
  float* __restrict__ wout = Wh + (size_t)(row0 + 8 * half) * OUT_F + col0 + idx;
  if (row0 + 16 <= n_nodes) {
#pragma unroll
    for (int i = 0; i < 8; ++i) wout[(size_t)i * OUT_F] = acc[i];
  } else {
#pragma unroll
    for (int i = 0; i < 8; ++i) {
      if (row0 + i + 8 * half < n_nodes) wout[(size_t)i * OUT_F] = acc[i];
    }
  }
}

// ---------------------------------------------------------------------------
// 2) s1[row] = Wh[row,:] . a[:128] ; s2[row] = Wh[row,:] . a[128:256]
//    One wave32 per row; 4 floats per lane; shuffle reduce.
// ---------------------------------------------------------------------------
__global__ __launch_bounds__(256) void gat_s1s2(const float* __restrict__ Wh,
                                                const float* __restrict__ a,
                                                float* __restrict__ s1,
                                                float* __restrict__ s2,
                                                int n_nodes) {
  int row = blockIdx.x * (blockDim.x >> 5) + (threadIdx.x >> 5);
  if (row >= n_nodes) return;
  int lane = threadIdx.x & 31;
  const float* __restrict__ wr = Wh + (size_t)row * OUT_F;
  float p1 = 0.0f, p2 = 0.0f;
#pragma unroll
  for (int c = lane; c < OUT_F; c += 32) {
    float w = wr[c];
    p1 += w * a[c];
    p2 += w * a[OUT_F + c];
  }
#pragma unroll
  for (int off = 16; off > 0; off >>= 1) {
    p1 += __shfl_down(p1, off, 32);
    p2 += __shfl_down(p2, off, 32);
  }
  if (lane == 0) { s1[row] = p1; s2[row] = p2; }
}

// ---------------------------------------------------------------------------
// 3) per-edge logit + segment max over src (atomic max on ordered uint)
// ---------------------------------------------------------------------------
__global__ __launch_bounds__(256) void gat_edge_logit(const long long* __restrict__ ei,
                                                      const float* __restrict__ s1,
                                                      const float* __restrict__ s2,
                                                      float* __restrict__ logits,
                                                      unsigned* __restrict__ segmax,
                                                      int n_edges) {
  int e = blockIdx.x * blockDim.x + threadIdx.x;
  if (e >= n_edges) return;
  int src = (int)ei[e];
  int dst = (int)ei[(size_t)n_edges + e];
  float x = s1[src] + s2[dst];
  x = (x >= 0.0f) ? x : NEG_SLOPE * x;     // leaky relu
  x *= INV_TEMP;                           // / temperature
  logits[e] = x;
  atomicMax(segmax + src, f2ord(x));
}

// ---------------------------------------------------------------------------
// 4) ex = exp(logit - segmax[src]); denom[src] += ex (atomic)
// ---------------------------------------------------------------------------
__global__ __launch_bounds__(256) void gat_edge_exp(const long long* __restrict__ ei,
                                                    const unsigned* __restrict__ segmax,
                                                    float* __restrict__ logits_ex,
                                                    float* __restrict__ denom,
                                                    int n_edges) {
  int e = blockIdx.x * blockDim.x + threadIdx.x;
  if (e >= n_edges) return;
  int src = (int)ei[e];
  float m = ord2f(segmax[src]);
  float ex = __expf(logits_ex[e] - m);
  logits_ex[e] = ex;                       // overwrite logits with ex
  atomAddF32(denom + src, ex);
}

// ---------------------------------------------------------------------------
// 5) out[src,:] += (ex/denom[src]) * Wh[dst,:]   — one wave32 per edge,
//    float4 per lane (32*4 = 128 features), 4 native f32 atomics per lane.
// ---------------------------------------------------------------------------
__global__ __launch_bounds__(256) void gat_aggregate(const long long* __restrict__ ei,
                                                     const float* __restrict__ ex,
                                                     const float* __restrict__ denom,
                                                     const float* __restrict__ Wh,
                                                     float* __restrict__ out,
                                                     int n_edges) {
  int e = blockIdx.x * (blockDim.x >> 5) + (threadIdx.x >> 5);
  if (e >= n_edges) return;
  int lane = threadIdx.x & 31;
  int src = (int)ei[e];
  int dst = (int)ei[(size_t)n_edges + e];
  float att = __fdividef(ex[e], denom[src]);
  float4 v = ((const float4*)(Wh + (size_t)dst * OUT_F))[lane];
  float* o = out + (size_t)src * OUT_F + lane * 4;
  atomAddF32(o + 0, att * v.x);
  atomAddF32(o + 1, att * v.y);
  atomAddF32(o + 2, att * v.z);
  atomAddF32(o + 3, att * v.w);
}

// ---------------------------------------------------------------------------
// 6) relu in place
// ---------------------------------------------------------------------------
__global__ __launch_bounds__(256) void gat_relu(float* __restrict__ out, int total) {
  int i = blockIdx.x * blockDim.x + threadIdx.x;
  if (i < total) {
    float v = out[i];
    out[i] = v > 0.0f ? v : 0.0f;
  }
}

// ---------------------------------------------------------------------------
static inline size_t align256(size_t x) { return (x + 255) & ~(size_t)255; }

extern "C" void kernel_launch(void* const* d_in, const int* in_sizes, int n_in,
                              void* d_out, int out_size, void* d_ws, size_t ws_size,
                              hipStream_t stream) {
  const float*     h  = (const float*)d_in[0];       // (N, 512)
  const float*     W  = (const float*)d_in[1];       // (512, 128)
  const float*     a  = (const float*)d_in[2];       // (256, 1)
  const long long* ei = (const long long*)d_in[3];   // (2, E) int64

  const int N = in_sizes[0] / IN_F;
  const int E = in_sizes[3] / 2;
  float* out = (float*)d_out;                        // (N, 128)

  // workspace carve-out
  char* ws = (char*)d_ws;
  size_t off = 0;
  float*    Wh     = (float*)(ws + off); off += align256((size_t)N * OUT_F * 4);
  float*    s1     = (float*)(ws + off); off += align256((size_t)N * 4);
  float*    s2     = (float*)(ws + off); off += align256((size_t)N * 4);
  unsigned* segmax = (unsigned*)(ws + off); off += align256((size_t)N * 4);
  float*    denom  = (float*)(ws + off); off += align256((size_t)N * 4);
  float*    exbuf  = (float*)(ws + off); off += align256((size_t)E * 4);
  (void)ws_size; (void)n_in; (void)out_size;

  // 0) init accumulators
  {
    int total = N * OUT_F;
    gat_init<<<(total + 255) / 256, 256, 0, stream>>>(out, segmax, denom, N);
  }
  // 1) Wh = h @ W   (WMMA fp32)
  gat_gemm_wh<<<(N + 15) / 16, 256, 0, stream>>>(h, W, Wh, N);
  // 2) s1, s2
  gat_s1s2<<<(N + 7) / 8, 256, 0, stream>>>(Wh, a, s1, s2, N);
  // 3) edge logits + segment max
  gat_edge_logit<<<(E + 255) / 256, 256, 0, stream>>>(ei, s1, s2, exbuf, segmax, E);
  // 4) exp + denom
  gat_edge_exp<<<(E + 255) / 256, 256, 0, stream>>>(ei, segmax, exbuf, denom, E);
  // 5) weighted scatter-add aggregation (wave per edge)
  gat_aggregate<<<(E + 7) / 8, 256, 0, stream>>>(ei, exbuf, denom, Wh, out, E);
  // 6) relu
  {
    int total = N * OUT_F;
    gat_relu<<<(total + 255) / 256, 256, 0, stream>>>(out, total);
  }
}